// LinearAttnLast_37752762532671
// MI455X (gfx1250) — compile-verified
//
#include <hip/hip_runtime.h>
#include <math.h>

typedef __attribute__((ext_vector_type(2))) float v2f;
typedef __attribute__((ext_vector_type(8))) float v8f;

#define LDSS 65   // padded row stride (floats) for a 16x64 tile in LDS

static __device__ __forceinline__ v8f wmma_f32_k4(v2f a, v2f b, v8f c) {
  // D = A(16x4) * B(4x16) + C(16x16), all fp32.  8-arg VOP3P form:
  // (neg_a, A, neg_b, B, c_mod, C, reuse_a, reuse_b)
  return __builtin_amdgcn_wmma_f32_16x16x4_f32(false, a, false, b, (short)0, c,
                                               false, false);
}

static __device__ __forceinline__ v8f v8f_zero() {
  v8f z = {0.f, 0.f, 0.f, 0.f, 0.f, 0.f, 0.f, 0.f};
  return z;
}

// ---------------------------------------------------------------------------
// Kernel 1: LayerNorm + QKV GEMM.  One wave = one 16-node tile.
//   x = LN(node_feat) ; qkv = x @ qkv_w(64x192) + qkv_b  -> (N,192)
// ---------------------------------------------------------------------------
__global__ __launch_bounds__(256) void k_ln_qkv(
    const float* __restrict__ node_feat, const float* __restrict__ ln_g,
    const float* __restrict__ ln_b, const float* __restrict__ qkv_w,
    const float* __restrict__ qkv_b, float* __restrict__ qkv, int N) {
  __shared__ float sx[8][16 * LDSS];
  const int lane = threadIdx.x & 31;
  const int w    = threadIdx.x >> 5;
  const int m    = lane & 15;
  const int hi   = lane >> 4;
  const int tile = blockIdx.x * 8 + w;
  const int base = tile * 16;

  // Phase 1: coalesced load of the 16x64 tile into padded LDS.
#pragma unroll
  for (int i = 0; i < 8; ++i) {
    const int rl = i * 2 + hi;      // local row 0..15
    const int c  = m * 4;           // column 0..60
    const int gr = base + rl;
    float4 val = make_float4(0.f, 0.f, 0.f, 0.f);
    if (gr < N) val = *(const float4*)(node_feat + (size_t)gr * 64 + c);
    float* dst = &sx[w][rl * LDSS + c];
    dst[0] = val.x; dst[1] = val.y; dst[2] = val.z; dst[3] = val.w;
  }
  __syncthreads();

  // Phase 2: LayerNorm. Lane pair (m, m+16) owns row m (32 cols each).
  float xv[32];
  float s = 0.f, sq = 0.f;
#pragma unroll
  for (int j = 0; j < 32; ++j) {
    const float x = sx[w][m * LDSS + hi * 32 + j];
    xv[j] = x; s += x; sq += x * x;
  }
  s  += __shfl_xor(s, 16, 32);
  sq += __shfl_xor(sq, 16, 32);
  const float mu  = s * (1.f / 64.f);
  const float var = sq * (1.f / 64.f) - mu * mu;
  const float inv = rsqrtf(var + 1e-5f);
#pragma unroll
  for (int j = 0; j < 32; ++j) {
    const int c = hi * 32 + j;
    sx[w][m * LDSS + c] = (xv[j] - mu) * inv * ln_g[c] + ln_b[c];
  }
  __syncthreads();

  // Phase 3: 16x192 GEMM via WMMA f32 16x16x4 (12 column tiles, K=64).
#pragma unroll 1
  for (int nt = 0; nt < 12; ++nt) {
    const int n = nt * 16 + m;
    v8f acc = v8f_zero();
#pragma unroll
    for (int k0 = 0; k0 < 64; k0 += 4) {
      const int kk = k0 + 2 * hi;
      v2f a; a.x = sx[w][m * LDSS + kk];  a.y = sx[w][m * LDSS + kk + 1];
      v2f b; b.x = qkv_w[kk * 192 + n];   b.y = qkv_w[(kk + 1) * 192 + n];
      acc = wmma_f32_k4(a, b, acc);
    }
    const float bias = qkv_b[n];
#pragma unroll
    for (int r = 0; r < 8; ++r) {
      const int gr = base + r + 8 * hi;   // D row = r + 8*(lane>>4)
      if (gr < N) qkv[(size_t)gr * 192 + n] = acc[r] + bias;
    }
  }
}

// ---------------------------------------------------------------------------
// Kernel 2: iv = node_vec(3N x 64) @ vec_w(64x128); input_dot += lo*hi per row
// ---------------------------------------------------------------------------
__global__ __launch_bounds__(256) void k_vec_dot(
    const float* __restrict__ node_vec, const float* __restrict__ vec_w,
    float* __restrict__ input_dot, int N) {
  __shared__ float sx[8][16 * LDSS];
  const int lane = threadIdx.x & 31;
  const int w    = threadIdx.x >> 5;
  const int m    = lane & 15;
  const int hi   = lane >> 4;
  const int rows3 = N * 3;
  const int base  = (blockIdx.x * 8 + w) * 16;

#pragma unroll
  for (int i = 0; i < 8; ++i) {
    const int rl = i * 2 + hi;
    const int c  = m * 4;
    const int gr = base + rl;
    float4 val = make_float4(0.f, 0.f, 0.f, 0.f);
    if (gr < rows3) val = *(const float4*)(node_vec + (size_t)gr * 64 + c);
    float* dst = &sx[w][rl * LDSS + c];
    dst[0] = val.x; dst[1] = val.y; dst[2] = val.z; dst[3] = val.w;
  }
  __syncthreads();

  v8f acc[8];
#pragma unroll
  for (int nt = 0; nt < 8; ++nt) acc[nt] = v8f_zero();

#pragma unroll 1
  for (int k0 = 0; k0 < 64; k0 += 4) {
    const int kk = k0 + 2 * hi;
    v2f a; a.x = sx[w][m * LDSS + kk];  a.y = sx[w][m * LDSS + kk + 1];
#pragma unroll
    for (int nt = 0; nt < 8; ++nt) {
      const int n = nt * 16 + m;
      v2f b; b.x = vec_w[kk * 128 + n]; b.y = vec_w[(kk + 1) * 128 + n];
      acc[nt] = wmma_f32_k4(a, b, acc[nt]);
    }
  }

  // input_dot[row/3][col] += iv_lo * iv_hi   (scatter over the 3-axis)
#pragma unroll
  for (int ct = 0; ct < 4; ++ct) {
    const int col = ct * 16 + m;
#pragma unroll
    for (int r = 0; r < 8; ++r) {
      const int gr = base + r + 8 * hi;
      if (gr < rows3) {
        const float p = acc[ct][r] * acc[ct + 4][r];
        atomicAdd(input_dot + (size_t)(gr / 3) * 64 + col, p);
      }
    }
  }
}

// ---------------------------------------------------------------------------
// Kernel 3: edge pass. One thread per (edge, head).
// ---------------------------------------------------------------------------
__global__ __launch_bounds__(256) void k_edge(
    const float* __restrict__ qkv, const float* __restrict__ edge_feat,
    const float* __restrict__ radial, const int* __restrict__ row,
    const int* __restrict__ col, float* __restrict__ m_agg, int E) {
  const int tid = blockIdx.x * 256 + threadIdx.x;
  if (tid >= E * 16) return;
  const int e = tid >> 4;
  const int h = tid & 15;
  const int r = row[e];
  const int c = col[e];
  const size_t rb = (size_t)r * 192 + h * 12;  // head chunk is 48B -> 16B aligned
  const size_t cb = (size_t)c * 192 + h * 12;
  const float4 q = *(const float4*)(qkv + rb);
  const float4 k = *(const float4*)(qkv + cb + 4);
  const float  d = q.x * k.x + q.y * k.y + q.z * k.z + q.w * k.w;
  // exact gelu: 0.5*x*(1+erf(x/sqrt(2)))
  const float  g = 0.5f * d * (1.f + erff(d * 0.70710678118654752f)) * radial[e];
  const float4 v  = *(const float4*)(qkv + cb + 8);
  const float4 ef = *(const float4*)(edge_feat + (size_t)tid * 4);
  float* dst = m_agg + (size_t)r * 64 + h * 4;
  atomicAdd(dst + 0, v.x * ef.x * g);
  atomicAdd(dst + 1, v.y * ef.y * g);
  atomicAdd(dst + 2, v.z * ef.z * g);
  atomicAdd(dst + 3, v.w * ef.w * g);
}

// ---------------------------------------------------------------------------
// Kernel 4: out = m_agg @ out_w(64x128) + out_b; y = input_dot*lo + hi
// ---------------------------------------------------------------------------
__global__ __launch_bounds__(256) void k_out(
    const float* __restrict__ m_agg, const float* __restrict__ out_w,
    const float* __restrict__ out_b, const float* __restrict__ input_dot,
    float* __restrict__ outp, int N) {
  __shared__ float sx[8][16 * LDSS];
  const int lane = threadIdx.x & 31;
  const int w    = threadIdx.x >> 5;
  const int m    = lane & 15;
  const int hi   = lane >> 4;
  const int base = (blockIdx.x * 8 + w) * 16;

#pragma unroll
  for (int i = 0; i < 8; ++i) {
    const int rl = i * 2 + hi;
    const int c  = m * 4;
    const int gr = base + rl;
    float4 val = make_float4(0.f, 0.f, 0.f, 0.f);
    if (gr < N) val = *(const float4*)(m_agg + (size_t)gr * 64 + c);
    float* dst = &sx[w][rl * LDSS + c];
    dst[0] = val.x; dst[1] = val.y; dst[2] = val.z; dst[3] = val.w;
  }
  __syncthreads();

  v8f acc[8];
#pragma unroll
  for (int nt = 0; nt < 8; ++nt) acc[nt] = v8f_zero();

#pragma unroll 1
  for (int k0 = 0; k0 < 64; k0 += 4) {
    const int kk = k0 + 2 * hi;
    v2f a; a.x = sx[w][m * LDSS + kk];  a.y = sx[w][m * LDSS + kk + 1];
#pragma unroll
    for (int nt = 0; nt < 8; ++nt) {
      const int n = nt * 16 + m;
      v2f b; b.x = out_w[kk * 128 + n]; b.y = out_w[(kk + 1) * 128 + n];
      acc[nt] = wmma_f32_k4(a, b, acc[nt]);
    }
  }

#pragma unroll
  for (int ct = 0; ct < 4; ++ct) {
    const int n  = ct * 16 + m;
    const float blo = out_b[n];
    const float bhi = out_b[64 + n];
#pragma unroll
    for (int r = 0; r < 8; ++r) {
      const int gr = base + r + 8 * hi;
      if (gr < N) {
        const float lo  = acc[ct][r] + blo;
        const float hv  = acc[ct + 4][r] + bhi;
        const float idv = input_dot[(size_t)gr * 64 + n];
        outp[(size_t)gr * 64 + n] = idv * lo + hv;
      }
    }
  }
}

// ---------------------------------------------------------------------------
extern "C" void kernel_launch(void* const* d_in, const int* in_sizes, int n_in,
                              void* d_out, int out_size, void* d_ws,
                              size_t ws_size, hipStream_t stream) {
  const float* node_feat = (const float*)d_in[0];
  const float* edge_feat = (const float*)d_in[1];
  const float* node_vec  = (const float*)d_in[2];
  const float* radial    = (const float*)d_in[3];
  const float* ln_g      = (const float*)d_in[4];
  const float* ln_b      = (const float*)d_in[5];
  const float* qkv_w     = (const float*)d_in[6];
  const float* qkv_b     = (const float*)d_in[7];
  const float* vec_w     = (const float*)d_in[8];
  const float* out_w     = (const float*)d_in[9];
  const float* out_b     = (const float*)d_in[10];
  const int*   row       = (const int*)d_in[11];
  const int*   col       = (const int*)d_in[12];
  // natoms (d_in[13]) is a device scalar; derive sizes on host:
  const int N = in_sizes[0] / 64;   // node_feat = (N, 64)
  const int E = in_sizes[3];        // radial    = (E,)

  float* qkv   = (float*)d_ws;                    // N*192
  float* idot  = qkv + (size_t)N * 192;           // N*64
  float* magg  = idot + (size_t)N * 64;           // N*64
  hipMemsetAsync(idot, 0, (size_t)N * 128 * sizeof(float), stream);

  const int tilesN = (N + 15) / 16;
  const int tiles3 = (3 * N + 15) / 16;
  k_ln_qkv<<<(tilesN + 7) / 8, 256, 0, stream>>>(node_feat, ln_g, ln_b, qkv_w,
                                                 qkv_b, qkv, N);
  k_vec_dot<<<(tiles3 + 7) / 8, 256, 0, stream>>>(node_vec, vec_w, idot, N);
  const long long tot = (long long)E * 16;
  k_edge<<<(int)((tot + 255) / 256), 256, 0, stream>>>(qkv, edge_feat, radial,
                                                       row, col, magg, E);
  k_out<<<(tilesN + 7) / 8, 256, 0, stream>>>(magg, out_w, out_b, idot,
                                              (float*)d_out, N);
}